// VGG16RoIHead_6356551598246
// MI455X (gfx1250) — compile-verified
//
#include <hip/hip_runtime.h>
#include <float.h>

// ---------------------------------------------------------------------------
// Problem constants (from the reference)
// ---------------------------------------------------------------------------
#define RT    128          // total ROIs (2*64)
#define CCH   512          // channels
#define HH    37
#define WW    37
#define PP    7
#define DFLAT (CCH * PP * PP)   // 25088
#define HID   4096
#define NLOC  84           // 21*4
#define NSCO  21

typedef float v2f __attribute__((ext_vector_type(2)));
typedef float v8f __attribute__((ext_vector_type(8)));

// ---------------------------------------------------------------------------
// RoIPool: one thread per output element of flat[RT, C*P*P]
// ---------------------------------------------------------------------------
__global__ __launch_bounds__(256) void roipool_kernel(
    const float* __restrict__ x,      // [2, C, H, W]
    const float* __restrict__ rois,   // [RT, 4] (x1,y1,x2,y2) image coords
    const int*   __restrict__ bidx,   // [RT]
    const int*   __restrict__ img_h_p,
    const int*   __restrict__ img_w_p,
    float*       __restrict__ flat)   // [RT, C*P*P]
{
    int idx = blockIdx.x * blockDim.x + threadIdx.x;
    if (idx >= RT * CCH * PP * PP) return;

    int pw = idx % PP;
    int t  = idx / PP;
    int ph = t % PP;
    t /= PP;
    int c = t % CCH;
    int r = t / CCH;

    float sx = (float)WW / (float)img_w_p[0];
    float sy = (float)HH / (float)img_h_p[0];

    const float* rr = rois + r * 4;
    // rnd(v) = floor(v*SCALE + 0.5), SCALE = 1
    int ws = (int)floorf(rr[0] * sx + 0.5f);
    int hs = (int)floorf(rr[1] * sy + 0.5f);
    int we = (int)floorf(rr[2] * sx + 0.5f);
    int he = (int)floorf(rr[3] * sy + 0.5f);

    float bw = fmaxf((float)(we - ws + 1), 1.0f) / (float)PP;
    float bh = fmaxf((float)(he - hs + 1), 1.0f) / (float)PP;

    int h0 = hs + (int)floorf((float)ph * bh);
    int h1 = hs + (int)ceilf(((float)ph + 1.0f) * bh);
    h0 = min(max(h0, 0), HH);
    h1 = min(max(h1, 0), HH);
    h1 = min(h1, h0 + PP);   // reference window limited to KH taps

    int w0 = ws + (int)floorf((float)pw * bw);
    int w1 = ws + (int)ceilf(((float)pw + 1.0f) * bw);
    w0 = min(max(w0, 0), WW);
    w1 = min(max(w1, 0), WW);
    w1 = min(w1, w0 + PP);

    int b = bidx[r];
    const float* xc = x + ((size_t)b * CCH + c) * (HH * WW);

    float m = -FLT_MAX;
    bool any = false;
    for (int h = h0; h < h1; ++h) {
        const float* row = xc + h * WW;
        for (int w = w0; w < w1; ++w) {
            m = fmaxf(m, row[w]);
            any = true;
        }
    }
    flat[idx] = any ? m : 0.0f;
}

// ---------------------------------------------------------------------------
// GEMM:  C[128, N] = act(A[128, K] @ W[N, K]^T + bias)
// grid.x = ceil(N/16) blocks, 256 threads = 8 waves; wave w owns M-tile w.
// All 8 waves of a block stream the same 16 W-rows (L0/L2 reuse) so the big
// weight matrix hits HBM once -> runs at the 23.3 TB/s bandwidth floor.
// Uses V_WMMA_F32_16X16X4_F32 (exact fp32, matches reference numerics).
// ---------------------------------------------------------------------------
__global__ __launch_bounds__(256) void gemm_wmma_f32(
    const float* __restrict__ A,     // [128, K] row-major
    const float* __restrict__ Wt,    // [N, K]  row-major (B = Wt^T)
    const float* __restrict__ bias,  // [N]
    float*       __restrict__ C,     // [128, N] row-major (ldc = N)
    int K, int N, int relu)
{
    const int lane  = threadIdx.x & 31;
    const int wave  = threadIdx.x >> 5;
    const int m0    = wave * 16;                 // M-tile base (M = 128 = 8*16)
    const int n0    = blockIdx.x * 16;           // N-tile base
    const int mrow  = lane & 15;                 // A row within tile
    const int khalf = (lane >> 4) * 2;           // K sub-offset: 0 or 2

    int ncol = n0 + (lane & 15);
    int nld  = (ncol < N) ? ncol : (N - 1);      // clamp load row (store is guarded)

    const float* aPtr = A  + (size_t)(m0 + mrow) * K + khalf;
    const float* bPtr = Wt + (size_t)nld * K + khalf;

    v8f acc = {0.f, 0.f, 0.f, 0.f, 0.f, 0.f, 0.f, 0.f};

#pragma unroll 8
    for (int k = 0; k < K; k += 4) {
        v2f a = *(const v2f*)(aPtr + k);   // A(m, k..k+1 | k+2..k+3)
        v2f b = *(const v2f*)(bPtr + k);   // B(k..k+1 | k+2..k+3, n)
        acc = __builtin_amdgcn_wmma_f32_16x16x4_f32(
            /*neg_a=*/false, a, /*neg_b=*/false, b,
            /*c_mod=*/(short)0, acc, /*reuse_a=*/false, /*reuse_b=*/false);
    }

    if (ncol < N) {
        float bv = bias[ncol];
        int rbase = m0 + ((lane >> 4) * 8);      // C rows: rbase + i
#pragma unroll
        for (int i = 0; i < 8; ++i) {
            float v = acc[i] + bv;
            if (relu) v = fmaxf(v, 0.0f);
            C[(size_t)(rbase + i) * N + ncol] = v;
        }
    }
}

// ---------------------------------------------------------------------------
// Host side
// ---------------------------------------------------------------------------
extern "C" void kernel_launch(void* const* d_in, const int* in_sizes, int n_in,
                              void* d_out, int out_size, void* d_ws, size_t ws_size,
                              hipStream_t stream) {
    (void)in_sizes; (void)n_in; (void)out_size; (void)ws_size;

    const float* x     = (const float*)d_in[0];
    const float* rois  = (const float*)d_in[1];
    const int*   ridx  = (const int*)  d_in[2];
    const int*   img_h = (const int*)  d_in[3];
    const int*   img_w = (const int*)  d_in[4];
    const float* W1    = (const float*)d_in[5];
    const float* b1    = (const float*)d_in[6];
    const float* W2    = (const float*)d_in[7];
    const float* b2    = (const float*)d_in[8];
    const float* Wc    = (const float*)d_in[9];
    const float* bc    = (const float*)d_in[10];
    const float* Ws    = (const float*)d_in[11];
    const float* bs    = (const float*)d_in[12];

    float* ws_f  = (float*)d_ws;
    float* flat  = ws_f;                         // [128, 25088]  (12.85 MB)
    float* fc6   = ws_f + (size_t)RT * DFLAT;    // [128, 4096]   ( 2 MB)
    float* fc7   = ws_f;                         // reuse flat's space (dead after fc6)

    float* locs   = (float*)d_out;               // [128, 84]
    float* scores = (float*)d_out + RT * NLOC;   // [128, 21]

    // 1) RoIPool -> flat
    {
        int total  = RT * CCH * PP * PP;
        int blocks = (total + 255) / 256;
        roipool_kernel<<<blocks, 256, 0, stream>>>(x, rois, ridx, img_h, img_w, flat);
    }
    // 2) fc6 = relu(flat @ W1^T + b1)    [128, 25088] x [25088, 4096]
    gemm_wmma_f32<<<HID / 16, 256, 0, stream>>>(flat, W1, b1, fc6, DFLAT, HID, 1);
    // 3) fc7 = relu(fc6 @ W2^T + b2)     [128, 4096] x [4096, 4096]
    gemm_wmma_f32<<<HID / 16, 256, 0, stream>>>(fc6, W2, b2, fc7, HID, HID, 1);
    // 4) heads (no relu), straight into d_out
    gemm_wmma_f32<<<(NLOC + 15) / 16, 256, 0, stream>>>(fc7, Wc, bc, locs,   HID, NLOC, 0);
    gemm_wmma_f32<<<(NSCO + 15) / 16, 256, 0, stream>>>(fc7, Ws, bs, scores, HID, NSCO, 0);
}